// AttnConv2d_58634893525304
// MI455X (gfx1250) — compile-verified
//
#include <hip/hip_runtime.h>
#include <math.h>

typedef float v2f __attribute__((ext_vector_type(2)));
typedef float v8f __attribute__((ext_vector_type(8)));

#define CIN   192
#define COUT  192
#define HH    96
#define WIMG  96
#define PIX   (HH * WIMG)     // 9216
#define KDIM  (CIN * 9)       // 1728
#define BATCH 8
#define LSUB  1024            // 32*32 non-overlapping patches

// f32 WMMA: D(16x16,f32) = A(16x4,f32) * B(4x16,f32) + C
__device__ __forceinline__ v8f wmma_f32(v2f a, v2f b, v8f c) {
    return __builtin_amdgcn_wmma_f32_16x16x4_f32(
        /*neg_a=*/false, a, /*neg_b=*/false, b,
        /*c_mod=*/(short)0, c, /*reuse_a=*/false, /*reuse_b=*/false);
}

__device__ __forceinline__ v2f lds_v2f(const float* p) {
    return *reinterpret_cast<const v2f*>(p);   // 8B-aligned by construction
}

// async 16B copy: global -> LDS, tracked on ASYNCcnt (CDNA5 async path)
__device__ __forceinline__ void async_copy_b128(unsigned lds_off,
                                                const float* gaddr) {
    asm volatile("global_load_async_to_lds_b128 %0, %1, off"
                 :: "v"(lds_off), "v"(gaddr) : "memory");
}
__device__ __forceinline__ void wait_async0() {
    asm volatile("s_wait_asynccnt 0x0" ::: "memory");
}

// ---------------------------------------------------------------------------
// Implicit-GEMM conv:  out[b, m, p] = sum_K w[b?][m, K] * im2col(x[b])[K, p]
// K = c*9 + kh*3 + kw ; p = h*96 + w ; pad=1, stride=1.
// Block: 256 thr = 8 waves (2x4), tile M=64 x N=128; each wave owns a 32x32
// sub-tile (4 f32 WMMA accumulators, fragments reused 2x each).
// Kblk = 16, double-buffered LDS, one barrier per panel.
// A panel staged with global_load_async_to_lds_b128 (ASYNCcnt);
// B panel (gathered, zero-padded im2col) staged through registers.
// wbs = per-batch weight stride (0 => shared weights).
// ---------------------------------------------------------------------------
__global__ __launch_bounds__(256)
void conv_gemm_wmma(const float* __restrict__ x, const float* __restrict__ w,
                    float* __restrict__ out, long long wbs)
{
    __shared__ float As[2][64][20];    // [buf][m][k] 80B rows: 16B-aligned quads
    __shared__ float Bs[2][16][129];   // [buf][k][n]

    const int t  = threadIdx.x;
    const int b  = blockIdx.z;
    const int m0 = blockIdx.y * 64;
    const int n0 = blockIdx.x * 128;
    const float* wp = w + (long long)b * wbs;
    const float* xb = x + (size_t)b * (CIN * PIX);

    const int lane = t & 31;
    const int wave = t >> 5;
    const int lo   = lane & 15;
    const int hi   = lane >> 4;
    const int wm   = wave >> 2;       // 0..1 : rows 32*wm
    const int wn   = wave & 3;        // 0..3 : cols 32*wn

    v8f acc[2][2];
    #pragma unroll
    for (int i = 0; i < 2; ++i)
        #pragma unroll
        for (int j = 0; j < 2; ++j) acc[i][j] = (v8f){0, 0, 0, 0, 0, 0, 0, 0};

    // A staging: thread -> (arow 0..63, ak in {0,4,8,12}); one async b128 each
    const int arow = t >> 2;
    const int ak   = (t & 3) * 4;
    const float* wr = wp + (size_t)(m0 + arow) * KDIM + ak;
    const unsigned ldsA0 = (unsigned)(size_t)&As[0][arow][ak];
    const unsigned ldsA1 = (unsigned)(size_t)&As[1][arow][ak];

    // B staging: thread -> (kB 0..15, nB 8-wide); incremental c/r state
    const int kB = t >> 4;
    const int nB = (t & 15) * 8;
    int cB = kB / 9;                 // computed once
    int rB = kB - cB * 9;
    const int pgl = n0 + nB;
    const int ph0 = pgl / WIMG;      // loop-invariant: 8-groups never cross a row
    const int pw0 = pgl - ph0 * WIMG;

    float bR[8];

    auto issue_A = [&](int bufi) {
        async_copy_b128(bufi ? ldsA1 : ldsA0, wr);
        wr += 16;
        __builtin_prefetch(wr, 0, 0);   // panel after next -> global_prefetch_b8
    };
    auto load_B = [&]() {
        int kh = (rB >= 6) ? 2 : ((rB >= 3) ? 1 : 0);
        int kw = rB - 3 * kh;
        int h  = ph0 + kh - 1;
        const float* src = xb + (size_t)cB * PIX + h * WIMG + (pw0 + kw - 1);
        bool hok = (h >= 0) && (h < HH);
        #pragma unroll
        for (int j = 0; j < 8; ++j) {
            int wq = pw0 + kw - 1 + j;
            bR[j] = (hok && wq >= 0 && wq < WIMG) ? src[j] : 0.0f;
        }
        // advance per-thread K-state to the next panel (replaces /9, /3)
        rB += 7; cB += 1;
        if (rB >= 9) { rB -= 9; cB += 1; }
    };
    auto store_B = [&](int bufi) {
        #pragma unroll
        for (int j = 0; j < 8; ++j) Bs[bufi][kB][nB + j] = bR[j];
    };

    issue_A(0);
    load_B();
    store_B(0);
    wait_async0();
    __syncthreads();

    int buf = 0;
    for (int kb = 0; kb < KDIM; kb += 16) {
        const bool more = (kb + 16) < KDIM;
        if (more) { issue_A(buf ^ 1); load_B(); }  // in flight during WMMA burst

        // ISA A(16x4) layout: v0 = K+0 (lanes 0-15) / K+2 (16-31), v1 = K+1/K+3
        #pragma unroll
        for (int kk = 0; kk < 16; kk += 4) {
            v2f a0 = lds_v2f(&As[buf][32 * wm + lo][kk + 2 * hi]);
            v2f a1 = lds_v2f(&As[buf][32 * wm + 16 + lo][kk + 2 * hi]);
            v2f b0, b1;
            b0.x = Bs[buf][kk + 2 * hi][32 * wn + lo];
            b0.y = Bs[buf][kk + 2 * hi + 1][32 * wn + lo];
            b1.x = Bs[buf][kk + 2 * hi][32 * wn + 16 + lo];
            b1.y = Bs[buf][kk + 2 * hi + 1][32 * wn + 16 + lo];
            acc[0][0] = wmma_f32(a0, b0, acc[0][0]);
            acc[0][1] = wmma_f32(a0, b1, acc[0][1]);
            acc[1][0] = wmma_f32(a1, b0, acc[1][0]);
            acc[1][1] = wmma_f32(a1, b1, acc[1][1]);
        }
        if (more) store_B(buf ^ 1);
        wait_async0();
        __syncthreads();
        buf ^= 1;
    }

    // C/D layout: vgpr r -> (M = r + 8*hi, N = lo)
    float* op = out + (size_t)b * (COUT * PIX);
    #pragma unroll
    for (int mi = 0; mi < 2; ++mi) {
        #pragma unroll
        for (int r = 0; r < 8; ++r) {
            int m = m0 + 32 * wm + 16 * mi + r + 8 * hi;
            size_t rowoff = (size_t)m * PIX + (n0 + 32 * wn + lo);
            op[rowoff]      = acc[mi][0][r];
            op[rowoff + 16] = acc[mi][1][r];
        }
    }
}

// ---------------------------------------------------------------------------
// attn[b,(ki,kj)][c,o] = sum over 32x32 subgrid (h=3ph+ki, w=3pw+kj) of
//                        y1[b,c,h,w] * y2[b,o,h,w]   (K = 1024)
// Written directly in final conv-weight layout attnW[b][o][c*9 + kpos].
// Also accumulates global sum / sumsq for the normalization pass.
// ---------------------------------------------------------------------------
__global__ __launch_bounds__(256)
void attn_gemm_wmma(const float* __restrict__ y1, const float* __restrict__ y2,
                    float* __restrict__ attnW, float* __restrict__ stats)
{
    __shared__ float As[16][65];   // [l][c]
    __shared__ float Bs[16][65];   // [l][o]
    __shared__ float redS[256];
    __shared__ float redQ[256];

    const int t    = threadIdx.x;
    const int bz   = blockIdx.z;          // b*9 + kpos
    const int b    = bz / 9;
    const int kpos = bz - b * 9;
    const int ki   = kpos / 3;
    const int kj   = kpos - ki * 3;
    const int c0   = blockIdx.y * 64;     // M = Cin
    const int o0   = blockIdx.x * 64;     // N = Cout
    const float* y1b = y1 + (size_t)b * (CIN * PIX);
    const float* y2b = y2 + (size_t)b * (COUT * PIX);

    const int lane = t & 31;
    const int wave = t >> 5;
    const int lo   = lane & 15;
    const int hi   = lane >> 4;
    const int wm   = wave >> 1;
    const int wn   = wave & 1;

    v8f acc0 = {0, 0, 0, 0, 0, 0, 0, 0};
    v8f acc1 = {0, 0, 0, 0, 0, 0, 0, 0};

    const int arow = t >> 2;          // c row 0..63
    const int al   = (t & 3) * 4;     // l 0,4,8,12
    const int lB   = t >> 4;          // l 0..15
    const int oB   = (t & 15) * 4;    // o 0..60

    for (int lb = 0; lb < LSUB; lb += 16) {
        #pragma unroll
        for (int j = 0; j < 4; ++j) {
            int l  = lb + al + j;
            int ph = l >> 5, pw = l & 31;
            As[al + j][arow] =
                y1b[(size_t)(c0 + arow) * PIX + (3 * ph + ki) * WIMG + (3 * pw + kj)];
        }
        {
            int l   = lb + lB;
            int ph  = l >> 5, pw = l & 31;
            int off = (3 * ph + ki) * WIMG + (3 * pw + kj);
            #pragma unroll
            for (int j = 0; j < 4; ++j)
                Bs[lB][oB + j] = y2b[(size_t)(o0 + oB + j) * PIX + off];
        }
        __syncthreads();

        #pragma unroll
        for (int kk = 0; kk < 16; kk += 4) {
            v2f a, b0, b1;
            a.x  = As[kk + 2 * hi][16 * wm + lo];
            a.y  = As[kk + 2 * hi + 1][16 * wm + lo];
            b0.x = Bs[kk + 2 * hi][32 * wn + lo];
            b0.y = Bs[kk + 2 * hi + 1][32 * wn + lo];
            b1.x = Bs[kk + 2 * hi][32 * wn + 16 + lo];
            b1.y = Bs[kk + 2 * hi + 1][32 * wn + 16 + lo];
            acc0 = wmma_f32(a, b0, acc0);
            acc1 = wmma_f32(a, b1, acc1);
        }
        __syncthreads();
    }

    float sum = 0.0f, sq = 0.0f;
    #pragma unroll
    for (int r = 0; r < 8; ++r) {
        int c = c0 + 16 * wm + r + 8 * hi;
        int o = o0 + 32 * wn + lo;
        float v0 = acc0[r], v1 = acc1[r];
        attnW[((size_t)b * COUT + o) * KDIM + c * 9 + kpos]        = v0;
        attnW[((size_t)b * COUT + (o + 16)) * KDIM + c * 9 + kpos] = v1;
        sum += v0 + v1;
        sq  += v0 * v0 + v1 * v1;
    }
    redS[t] = sum;
    redQ[t] = sq;
    __syncthreads();
    for (int off = 128; off > 0; off >>= 1) {
        if (t < off) { redS[t] += redS[t + off]; redQ[t] += redQ[t + off]; }
        __syncthreads();
    }
    if (t == 0) {
        atomicAdd(&stats[0], redS[0]);
        atomicAdd(&stats[1], redQ[0]);
    }
}

__global__ void zero_stats(float* s) { s[0] = 0.0f; s[1] = 0.0f; }

// attn = mom*attn + (attn - mean) / (std_unbiased + eps)
__global__ __launch_bounds__(256)
void attn_norm(float* __restrict__ attnW, const float* __restrict__ stats,
               const float* __restrict__ momp, int n)
{
    int i = blockIdx.x * blockDim.x + threadIdx.x;
    if (i >= n) return;
    float N    = (float)n;
    float mean = stats[0] / N;
    float var  = (stats[1] - N * mean * mean) / (N - 1.0f);
    float sdev = sqrtf(fmaxf(var, 0.0f));
    float inv  = 1.0f / (sdev + 1e-4f);
    float mom  = momp[0];
    float a    = attnW[i];
    attnW[i] = mom * a + (a - mean) * inv;
}

extern "C" void kernel_launch(void* const* d_in, const int* in_sizes, int n_in,
                              void* d_out, int out_size, void* d_ws, size_t ws_size,
                              hipStream_t stream)
{
    const float* x1      = (const float*)d_in[0];
    const float* x2      = (const float*)d_in[1];
    const float* key_w   = (const float*)d_in[2];
    const float* query_w = (const float*)d_in[3];
    const float* mom     = (const float*)d_in[4];
    float* out = (float*)d_out;

    float* y1    = (float*)d_ws;                          // B*CIN*PIX  floats
    float* attnW = y1 + (size_t)BATCH * CIN * PIX;        // B*COUT*KDIM floats
    float* stats = attnW + (size_t)BATCH * COUT * KDIM;   // 2 floats
    float* y2    = out;   // staged in d_out; consumed before final conv writes

    dim3 cgrid(PIX / 128, COUT / 64, BATCH);              // (72, 3, 8)

    zero_stats<<<1, 1, 0, stream>>>(stats);
    conv_gemm_wmma<<<cgrid, 256, 0, stream>>>(x1, key_w,   y1, 0LL);
    conv_gemm_wmma<<<cgrid, 256, 0, stream>>>(x2, query_w, y2, 0LL);
    attn_gemm_wmma<<<dim3(COUT / 64, CIN / 64, BATCH * 9), 256, 0, stream>>>(
        y1, y2, attnW, stats);
    int n = BATCH * COUT * KDIM;
    attn_norm<<<(n + 255) / 256, 256, 0, stream>>>(attnW, stats, mom, n);
    conv_gemm_wmma<<<cgrid, 256, 0, stream>>>(x1, attnW, out,
                                              (long long)COUT * KDIM);
}